// EBMDocking_36584531427658
// MI455X (gfx1250) — compile-verified
//
#include <hip/hip_runtime.h>
#include <math.h>

#define NB 2
#define NIMG 100
#define NN_ (NIMG*NIMG)
#define SP 200                 // true DFT size
#define SPP 208                // padded to 13 WMMA tiles
#define SSP (SPP*SPP)          // padded matrix elements
#define KK 200                 // contraction length (true size)
#define TM_ 13                 // 208/16 tiles per dim
#define TNP 7                  // ceil(13/2) N-tile pairs
#define RTOT 360
#define RC 8
#define NCHUNK (RTOT/RC)
#define EPSF 1e-12f
#define PI_F 3.14159265358979323846f

typedef float v2f __attribute__((ext_vector_type(2)));
typedef float v8f __attribute__((ext_vector_type(8)));

#if __has_builtin(__builtin_amdgcn_global_load_async_to_lds_b32)
#define HAVE_ASYNC_LDS 1
typedef __attribute__((address_space(1))) int gas_int;   // global (AS1)
typedef __attribute__((address_space(3))) int las_int;   // LDS (AS3)
#endif

// ---------------------------------------------------------------------------
// DFT twiddles, zero-padded to SPP x SPP: C[k,n]=cos(2pi*k*n/SP), S=sin.
// ---------------------------------------------------------------------------
__global__ void init_dft_kernel(float* Cm, float* Sm) {
  int idx = blockIdx.x * blockDim.x + threadIdx.x;
  if (idx >= SSP) return;
  int k = idx / SPP, n = idx % SPP;
  float cv = 0.f, sv = 0.f;
  if (k < SP && n < SP) {
    int p = (k * n) % SP;
    float ang = (2.0f * PI_F / SP) * (float)p;
    cv = cosf(ang); sv = sinf(ang);
  }
  Cm[idx] = cv; Sm[idx] = sv;
}

__global__ void init_state_kernel(unsigned* best, float* pos) {
  int b = threadIdx.x;
  if (b < NB) { best[b] = 0x007fffffu; /* enc(-inf) */ pos[b] = 0.f; }
}

// ---------------------------------------------------------------------------
// conv1 (1->9, 5x5, pad2) + norm-relu fields (0,1),(1,2),(3,2),(5,2),(7,2)
// si: 0,1 = receptor b ; 2,3 = ligand b
// ---------------------------------------------------------------------------
__global__ void conv1_norm_kernel(const float* __restrict__ rec,
                                  const float* __restrict__ lig,
                                  const float* __restrict__ w1,
                                  float* __restrict__ h1) {
  int idx = blockIdx.x * blockDim.x + threadIdx.x;
  if (idx >= 4 * NN_) return;
  int x = idx % NIMG, y = (idx / NIMG) % NIMG, si = idx / NN_;
  const float* img = (si < 2) ? (rec + (long)si * NN_) : (lig + (long)(si - 2) * NN_);
  float a[9];
#pragma unroll
  for (int o = 0; o < 9; ++o) {
    float s = 0.f;
    for (int ky = 0; ky < 5; ++ky) {
      int iy = y + ky - 2;
      if (iy < 0 || iy >= NIMG) continue;
      for (int kx = 0; kx < 5; ++kx) {
        int ix = x + kx - 2;
        if (ix < 0 || ix >= NIMG) continue;
        s += img[iy * NIMG + ix] * w1[o * 25 + ky * 5 + kx];
      }
    }
    a[o] = s;
  }
  float out[9];
  { float n = sqrtf(a[0] * a[0] + EPSF); out[0] = a[0] * (n / (n + EPSF)); }
#pragma unroll
  for (int g = 0; g < 4; ++g) {
    int st = 1 + g * 2;
    float n = sqrtf(a[st] * a[st] + a[st + 1] * a[st + 1] + EPSF);
    float f = n / (n + EPSF);
    out[st] = a[st] * f; out[st + 1] = a[st + 1] * f;
  }
  long base = (long)si * 9 * NN_ + y * NIMG + x;
#pragma unroll
  for (int o = 0; o < 9; ++o) h1[base + (long)o * NN_] = out[o];
}

// conv2 (9->3) + norm-relu (0,1),(1,2) + features |h0|, sqrt(h1^2+h2^2+eps)
__global__ void conv2_feat_kernel(const float* __restrict__ h1,
                                  const float* __restrict__ w2,
                                  float* __restrict__ feat) {
  int idx = blockIdx.x * blockDim.x + threadIdx.x;
  if (idx >= 4 * NN_) return;
  int x = idx % NIMG, y = (idx / NIMG) % NIMG, si = idx / NN_;
  const float* hin = h1 + (long)si * 9 * NN_;
  float a[3];
#pragma unroll
  for (int o = 0; o < 3; ++o) {
    float s = 0.f;
    for (int i = 0; i < 9; ++i) {
      const float* ch = hin + (long)i * NN_;
      for (int ky = 0; ky < 5; ++ky) {
        int iy = y + ky - 2;
        if (iy < 0 || iy >= NIMG) continue;
        for (int kx = 0; kx < 5; ++kx) {
          int ix = x + kx - 2;
          if (ix < 0 || ix >= NIMG) continue;
          s += ch[iy * NIMG + ix] * w2[(o * 9 + i) * 25 + ky * 5 + kx];
        }
      }
    }
    a[o] = s;
  }
  float n0 = sqrtf(a[0] * a[0] + EPSF);
  float h0 = a[0] * (n0 / (n0 + EPSF));
  float n1 = sqrtf(a[1] * a[1] + a[2] * a[2] + EPSF);
  float f1 = n1 / (n1 + EPSF);
  float hv1 = a[1] * f1, hv2 = a[2] * f1;
  long p = (long)y * NIMG + x;
  feat[((long)si * 2 + 0) * NN_ + p] = fabsf(h0);
  feat[((long)si * 2 + 1) * NN_ + p] = sqrtf(hv1 * hv1 + hv2 * hv2 + EPSF);
}

// receptor features zero-padded to SPPxSPP; img = b*2+ch
__global__ void recpad_kernel(const float* __restrict__ feat, float* __restrict__ recpad) {
  long idx = (long)blockIdx.x * blockDim.x + threadIdx.x;
  if (idx >= 4L * SSP) return;
  int x = (int)(idx % SPP), y = (int)((idx / SPP) % SPP);
  int img = (int)(idx / SSP);
  float v = 0.f;
  if (y < NIMG && x < NIMG) v = feat[(long)img * NN_ + y * NIMG + x];
  recpad[idx] = v;
}

// bilinear rotation of ligand features + zero pad; layout ((b*RC+lr)*2+ch)*SSP
__global__ void rotate_pad_kernel(const float* __restrict__ feat,
                                  float* __restrict__ rotpad, int r0) {
  long idx = (long)blockIdx.x * blockDim.x + threadIdx.x;
  if (idx >= (long)NB * RC * 2 * SSP) return;
  int x = (int)(idx % SPP);
  int y = (int)((idx / SPP) % SPP);
  int ch = (int)((idx / SSP) % 2);
  int lr = (int)((idx / ((long)SSP * 2)) % RC);
  int b  = (int)( idx / ((long)SSP * 2 * RC));
  float val = 0.f;
  if (y < NIMG && x < NIMG) {
    int r = r0 + lr;
    float theta = -PI_F + (float)r * (2.0f * PI_F / (float)RTOT);
    float ct = cosf(theta), st = sinf(theta);
    const float c = (NIMG - 1) * 0.5f;
    float xs = (float)x - c, ys = (float)y - c;
    float xq = ct * xs + st * ys + c;
    float yq = -st * xs + ct * ys + c;
    float x0f = floorf(xq), y0f = floorf(yq);
    float wx = xq - x0f, wy = yq - y0f;
    int x0 = (int)x0f, y0 = (int)y0f;
    const float* img = feat + ((long)(2 + b) * 2 + ch) * NN_;
    auto g = [&](int yi, int xi) -> float {
      bool valid = (yi >= 0) && (yi < NIMG) && (xi >= 0) && (xi < NIMG);
      int yc = min(max(yi, 0), NIMG - 1), xc = min(max(xi, 0), NIMG - 1);
      return valid ? img[yc * NIMG + xc] : 0.f;
    };
    val = g(y0, x0) * (1.f - wy) * (1.f - wx) + g(y0, x0 + 1) * (1.f - wy) * wx +
          g(y0 + 1, x0) * wy * (1.f - wx) + g(y0 + 1, x0 + 1) * wy * wx;
  }
  rotpad[idx] = val;
}

// ---------------------------------------------------------------------------
// Fused two-term batched GEMM on fp32 WMMA over 208-padded matrices:
//   D[img] = s0*(A0[img]*B0) + s1*(A1[img]*B1)   (stride 0 = shared operand)
// 8 waves/block = 8 M-tiles; each wave computes TWO N-tiles from one A
// fragment (2 accumulators -> 2/4 WMMAs per A load). B slabs (200x16 each,
// K-pair interleaved) async-staged to 51KB of LDS. Software-pipelined K loop
// (prefetch k+4 while computing k). No bounds checks: padding is zero and
// stays zero, EXEC is all-1s throughout.
// ---------------------------------------------------------------------------
template <int TWO>
__global__ void __launch_bounds__(256) gemm2_wmma_kernel(
    const float* __restrict__ A0, long sA0, const float* __restrict__ B0, long sB0, float s0,
    const float* __restrict__ A1, long sA1, const float* __restrict__ B1, long sB1, float s1,
    float* __restrict__ D, long sD) {
  __shared__ __align__(16) float bs[2][2][KK * 16];   // [matrix][ntile][pairs]
  int img = blockIdx.y;
  int tnp = blockIdx.x % TNP;         // N-tile pair
  int mg  = blockIdx.x / TNP;         // 0,1 : M-tile group of 8
  int n0 = tnp * 32;
  const float* a0 = A0 + (long)img * sA0;
  const float* a1 = A1 + (long)img * sA1;
  const float* b0 = B0 + (long)img * sB0 + n0;
  const float* b1 = B1 + (long)img * sB1 + n0;
  int tid = threadIdx.x;

  // Cooperative staging of 4 B slabs (2 matrices x 2 N-tiles), K-pair
  // interleaved so each lane's (k,k+1) fragment is one 8-byte LDS read.
  // i = (k<<6) | (n<<2) | (mat<<1) | nt : 12800 elems / 256 thr = 50 iters.
  for (int i = tid; i < 4 * KK * 16; i += 256) {
    int nt = i & 1, mat = (i >> 1) & 1;
    int n = (i >> 2) & 15, k = i >> 6;
    const float* src = (mat ? b1 : b0) + (long)k * SPP + nt * 16 + n;
    float* dst = &bs[mat][nt][(((k >> 1) << 4) + n) * 2 + (k & 1)];
#ifdef HAVE_ASYNC_LDS
    __builtin_amdgcn_global_load_async_to_lds_b32((gas_int*)src, (las_int*)dst, 0, 0);
#else
    *dst = *src;
#endif
  }
#ifdef HAVE_ASYNC_LDS
#if __has_builtin(__builtin_amdgcn_s_wait_asynccnt)
  __builtin_amdgcn_s_wait_asynccnt(0);
#else
  asm volatile("s_wait_asynccnt 0x0" ::: "memory");
#endif
#endif
  __syncthreads();

  int wave = tid >> 5, lane = tid & 31;
  int tm = mg * 8 + wave;
  if (tm < TM_) {
    int m0 = tm * 16;
    int r = lane & 15;              // A: M row / B: N col within tile
    int kh = (lane >> 4) << 1;      // K-half select {0,1} -> {0,2}
    const float* ga0 = a0 + (long)(m0 + r) * SPP + kh;
    const float* ga1 = a1 + (long)(m0 + r) * SPP + kh;
    int lofs = (((kh >> 1) << 4) + r) * 2;
    const float* lb00 = &bs[0][0][lofs];
    const float* lb01 = &bs[0][1][lofs];
    const float* lb10 = &bs[1][0][lofs];
    const float* lb11 = &bs[1][1][lofs];
    v8f acc0 = {0.f, 0.f, 0.f, 0.f, 0.f, 0.f, 0.f, 0.f};
    v8f acc1 = acc0;
    // pipeline prologue: fragments for k0 = 0
    v2f a0c = *(const v2f*)(ga0);
    v2f b00c = *(const v2f*)(lb00);
    v2f b01c = *(const v2f*)(lb01);
    v2f a1c{}, b10c{}, b11c{};
    if (TWO) {
      a1c = *(const v2f*)(ga1);
      b10c = *(const v2f*)(lb10);
      b11c = *(const v2f*)(lb11);
    }
    for (int k0 = 0; k0 < KK - 4; k0 += 4) {
      int kn = k0 + 4;
      // issue next-step loads before consuming current fragments
      v2f a0n = *(const v2f*)(ga0 + kn);
      v2f b00n = *(const v2f*)(lb00 + (kn << 4));
      v2f b01n = *(const v2f*)(lb01 + (kn << 4));
      v2f a1n{}, b10n{}, b11n{};
      if (TWO) {
        a1n = *(const v2f*)(ga1 + kn);
        b10n = *(const v2f*)(lb10 + (kn << 4));
        b11n = *(const v2f*)(lb11 + (kn << 4));
      }
      v2f as0 = a0c * s0;
      acc0 = __builtin_amdgcn_wmma_f32_16x16x4_f32(false, as0, false, b00c, (short)0, acc0,
                                                   false, false);
      acc1 = __builtin_amdgcn_wmma_f32_16x16x4_f32(false, as0, false, b01c, (short)0, acc1,
                                                   false, false);
      if (TWO) {
        v2f as1 = a1c * s1;
        acc0 = __builtin_amdgcn_wmma_f32_16x16x4_f32(false, as1, false, b10c, (short)0, acc0,
                                                     false, false);
        acc1 = __builtin_amdgcn_wmma_f32_16x16x4_f32(false, as1, false, b11c, (short)0, acc1,
                                                     false, false);
      }
      a0c = a0n; b00c = b00n; b01c = b01n;
      if (TWO) { a1c = a1n; b10c = b10n; b11c = b11n; }
    }
    // epilogue: last k-step
    {
      v2f as0 = a0c * s0;
      acc0 = __builtin_amdgcn_wmma_f32_16x16x4_f32(false, as0, false, b00c, (short)0, acc0,
                                                   false, false);
      acc1 = __builtin_amdgcn_wmma_f32_16x16x4_f32(false, as0, false, b01c, (short)0, acc1,
                                                   false, false);
      if (TWO) {
        v2f as1 = a1c * s1;
        acc0 = __builtin_amdgcn_wmma_f32_16x16x4_f32(false, as1, false, b10c, (short)0, acc0,
                                                     false, false);
        acc1 = __builtin_amdgcn_wmma_f32_16x16x4_f32(false, as1, false, b11c, (short)0, acc1,
                                                     false, false);
      }
    }
    float* dp = D + (long)img * sD + (long)m0 * SPP + n0;
    int colD = lane & 15;
    int rb = (lane >> 4) * 8;
#pragma unroll
    for (int j = 0; j < 8; ++j) dp[(long)(rb + j) * SPP + colD] = acc0[j];
    if (n0 + 16 < SPP) {              // second N-tile exists (tnp < 6)
#pragma unroll
      for (int j = 0; j < 8; ++j) dp[(long)(rb + j) * SPP + 16 + colD] = acc1[j];
    }
  }
}

// freq combine: sf = wb*conj(rb)*lb + w1*conj(rb)*lB + w2*conj(rB)*lb - wk*conj(rB)*lB
__global__ void combine_kernel(const float* __restrict__ recZr, const float* __restrict__ recZi,
                               const float* __restrict__ ligZr, const float* __restrict__ ligZi,
                               float* __restrict__ sfr, float* __restrict__ sfi,
                               const float* __restrict__ wB, const float* __restrict__ w1,
                               const float* __restrict__ w2, const float* __restrict__ wK) {
  long idx = (long)blockIdx.x * blockDim.x + threadIdx.x;
  if (idx >= (long)NB * RC * SSP) return;
  int e = (int)(idx % SSP);
  int lr = (int)((idx / SSP) % RC);
  int b = (int)(idx / ((long)SSP * RC));
  float rbr = recZr[((long)b * 2 + 0) * SSP + e], rbi = recZi[((long)b * 2 + 0) * SSP + e];
  float rBr = recZr[((long)b * 2 + 1) * SSP + e], rBi = recZi[((long)b * 2 + 1) * SSP + e];
  long lbase = ((long)(b * RC + lr) * 2) * SSP + e;
  float lbr = ligZr[lbase], lbi = ligZi[lbase];
  float lBr = ligZr[lbase + SSP], lBi = ligZi[lbase + SSP];
  float t1r = rbr * lbr + rbi * lbi, t1i = rbr * lbi - rbi * lbr;
  float t2r = rbr * lBr + rbi * lBi, t2i = rbr * lBi - rbi * lBr;
  float t3r = rBr * lbr + rBi * lbi, t3i = rBr * lbi - rBi * lbr;
  float t4r = rBr * lBr + rBi * lBi, t4i = rBr * lBi - rBi * lBr;
  float cb = wB[0], c1 = w1[0], c2 = w2[0], ck = wK[0];
  long o = (long)(b * RC + lr) * SSP + e;
  sfr[o] = cb * t1r + c1 * t2r + c2 * t3r - ck * t4r;
  sfi[o] = cb * t1i + c1 * t2i + c2 * t3i - ck * t4i;
}

__device__ inline unsigned fenc(float f) {
  int i = __float_as_int(f);
  return (i < 0) ? ~((unsigned)i) : ((unsigned)i | 0x80000000u);
}

// max over the true 200x200 region of each padded score slab
__global__ void reduce_max_kernel(const float* __restrict__ scores, unsigned* __restrict__ best) {
  int img = blockIdx.x;               // NB*RC slabs
  int b = img / RC;
  const float* s = scores + (long)img * SSP;
  float m = -3.4e38f;
  for (int i = threadIdx.x; i < SP * SP; i += blockDim.x) {
    int y = i / SP, x = i % SP;
    m = fmaxf(m, s[(long)y * SPP + x]);
  }
  __shared__ float sm[256];
  sm[threadIdx.x] = m;
  __syncthreads();
  for (int off = 128; off > 0; off >>= 1) {
    if ((int)threadIdx.x < off) sm[threadIdx.x] = fmaxf(sm[threadIdx.x], sm[threadIdx.x + off]);
    __syncthreads();
  }
  if (threadIdx.x == 0) atomicMax(&best[b], fenc(sm[0]));
}

__global__ void gather_pos_kernel(const float* __restrict__ scores,
                                  const int* __restrict__ gt_rot,
                                  const int* __restrict__ gt_txy,
                                  float* __restrict__ pos, int r0) {
  int b = threadIdx.x;
  if (b >= NB) return;
  int g = gt_rot[b];
  if (g >= r0 && g < r0 + RC) {
    int ty = gt_txy[b * 2 + 0], tx = gt_txy[b * 2 + 1];
    pos[b] = scores[(long)(b * RC + (g - r0)) * SSP + (long)ty * SPP + tx];
  }
}

__global__ void final_kernel(const unsigned* __restrict__ best,
                             const float* __restrict__ pos, float* __restrict__ out) {
  if (threadIdx.x != 0 || blockIdx.x != 0) return;
  float Lp = 0.f, Ln = 0.f;
  for (int b = 0; b < NB; ++b) {
    float p = pos[b];
    Lp += p + p * p;
    unsigned u = best[b];
    int i = (u & 0x80000000u) ? (int)(u & 0x7fffffffu) : (int)~u;
    float m = __int_as_float(i);
    Ln += -m + m * m;
  }
  out[0] = Lp / NB;
  out[1] = Ln / NB;
}

// ---------------------------------------------------------------------------
static void gemm2(hipStream_t st, const float* A0, long sA0, const float* B0, long sB0,
                  float s0, const float* A1, long sA1, const float* B1, long sB1, float s1,
                  float* D, long sD, int nb, int two) {
  dim3 grid(2 * TNP, nb, 1);          // 2 M-groups x 7 N-tile pairs
  if (two)
    gemm2_wmma_kernel<1><<<grid, dim3(256), 0, st>>>(A0, sA0, B0, sB0, s0, A1, sA1, B1, sB1,
                                                     s1, D, sD);
  else
    gemm2_wmma_kernel<0><<<grid, dim3(256), 0, st>>>(A0, sA0, B0, sB0, s0, A1, sA1, B1, sB1,
                                                     s1, D, sD);
}

extern "C" void kernel_launch(void* const* d_in, const int* in_sizes, int n_in,
                              void* d_out, int out_size, void* d_ws, size_t ws_size,
                              hipStream_t stream) {
  const float* receptor = (const float*)d_in[0];
  const float* ligand   = (const float*)d_in[1];
  const float* w1p      = (const float*)d_in[2];
  const float* w2p      = (const float*)d_in[3];
  const float* wB       = (const float*)d_in[4];
  const float* wc1      = (const float*)d_in[5];
  const float* wc2      = (const float*)d_in[6];
  const float* wK       = (const float*)d_in[7];
  const int*   gt_rot   = (const int*)d_in[8];
  const int*   gt_txy   = (const int*)d_in[9];

  float* w = (float*)d_ws;
  size_t o = 0;
  auto alloc = [&](size_t n) { float* p = w + o; o += n; return p; };
  float* Cm     = alloc(SSP);
  float* Sm     = alloc(SSP);
  float* h1b    = alloc((size_t)4 * 9 * NN_);
  float* feat   = alloc((size_t)4 * 2 * NN_);
  float* recpad = alloc((size_t)4 * SSP);
  float* recYr  = alloc((size_t)4 * SSP);
  float* recYi  = alloc((size_t)4 * SSP);
  float* recZr  = alloc((size_t)4 * SSP);
  float* recZi  = alloc((size_t)4 * SSP);
  const int LC = NB * RC * 2;          // 32 lig images per chunk
  float* rotpad = alloc((size_t)LC * SSP);
  float* lYr    = alloc((size_t)LC * SSP);
  float* lYi    = alloc((size_t)LC * SSP);
  float* lZr    = alloc((size_t)LC * SSP);
  float* lZi    = alloc((size_t)LC * SSP);
  float* sfr    = alloc((size_t)NB * RC * SSP);
  float* sfi    = alloc((size_t)NB * RC * SSP);
  float* Trb    = alloc((size_t)NB * RC * SSP);
  float* Tib    = alloc((size_t)NB * RC * SSP);
  float* sc     = alloc((size_t)NB * RC * SSP);
  unsigned* best = (unsigned*)alloc(2);
  float* pos    = alloc(2);

  init_dft_kernel<<<(SSP + 255) / 256, 256, 0, stream>>>(Cm, Sm);
  init_state_kernel<<<1, 32, 0, stream>>>(best, pos);

  conv1_norm_kernel<<<(4 * NN_ + 255) / 256, 256, 0, stream>>>(receptor, ligand, w1p, h1b);
  conv2_feat_kernel<<<(4 * NN_ + 255) / 256, 256, 0, stream>>>(h1b, w2p, feat);
  recpad_kernel<<<(4 * SSP + 255) / 256, 256, 0, stream>>>(feat, recpad);

  // receptor forward DFT (4 images): Y = X*(C - iS); Z = (C - iS)*Y
  gemm2(stream, recpad, SSP, Cm, 0, 1.f, recpad, SSP, Cm, 0, 0.f, recYr, SSP, 4, 0);
  gemm2(stream, recpad, SSP, Sm, 0, -1.f, recpad, SSP, Sm, 0, 0.f, recYi, SSP, 4, 0);
  gemm2(stream, Cm, 0, recYr, SSP, 1.f, Sm, 0, recYi, SSP, 1.f, recZr, SSP, 4, 1);
  gemm2(stream, Cm, 0, recYi, SSP, 1.f, Sm, 0, recYr, SSP, -1.f, recZi, SSP, 4, 1);

  const float INV = 1.0f / (float)(SP * SP);
  for (int c = 0; c < NCHUNK; ++c) {
    int r0 = c * RC;
    rotate_pad_kernel<<<(int)(((long)LC * SSP + 255) / 256), 256, 0, stream>>>(feat, rotpad, r0);
    // forward DFT of rotated ligand chunk
    gemm2(stream, rotpad, SSP, Cm, 0, 1.f, rotpad, SSP, Cm, 0, 0.f, lYr, SSP, LC, 0);
    gemm2(stream, rotpad, SSP, Sm, 0, -1.f, rotpad, SSP, Sm, 0, 0.f, lYi, SSP, LC, 0);
    gemm2(stream, Cm, 0, lYr, SSP, 1.f, Sm, 0, lYi, SSP, 1.f, lZr, SSP, LC, 1);
    gemm2(stream, Cm, 0, lYi, SSP, 1.f, Sm, 0, lYr, SSP, -1.f, lZi, SSP, LC, 1);
    // combine 4 weighted conj-products
    combine_kernel<<<(int)(((long)NB * RC * SSP + 255) / 256), 256, 0, stream>>>(
        recZr, recZi, lZr, lZi, sfr, sfi, wB, wc1, wc2, wK);
    // inverse DFT -> real scores (T = (C + iS)*SF ; s = Re(T*(C + iS))/S^2)
    gemm2(stream, Cm, 0, sfr, SSP, 1.f, Sm, 0, sfi, SSP, -1.f, Trb, SSP, NB * RC, 1);
    gemm2(stream, Cm, 0, sfi, SSP, 1.f, Sm, 0, sfr, SSP, 1.f, Tib, SSP, NB * RC, 1);
    gemm2(stream, Trb, SSP, Cm, 0, INV, Tib, SSP, Sm, 0, -INV, sc, SSP, NB * RC, 1);
    reduce_max_kernel<<<NB * RC, 256, 0, stream>>>(sc, best);
    gather_pos_kernel<<<1, 32, 0, stream>>>(sc, gt_rot, gt_txy, pos, r0);
  }
  final_kernel<<<1, 32, 0, stream>>>(best, pos, (float*)d_out);
}